// diy_attention_15831249453460
// MI455X (gfx1250) — compile-verified
//
#include <hip/hip_runtime.h>
#include <math.h>

// Problem constants (reference: B=2048, T=50, D=256)
#define BB 2048
#define TT 50
#define DD 256
#define NEGV -100000.0f

typedef float v2f __attribute__((ext_vector_type(2)));
typedef float v8f __attribute__((ext_vector_type(8)));

// ---------------------------------------------------------------------------
// Kernel 1: q[b,:] = (x[b,T-1,:] * td[b,T-1]) @ Wq^T + bq ; masked-fill NEG
// GEMM M=2048 (batch), N=256, K=256 via V_WMMA_F32_16X16X4_F32.
// One 16x16 tile per wave; 8 waves / block; 2048 tiles total -> 256 blocks.
// ---------------------------------------------------------------------------
__global__ __launch_bounds__(256) void k_qproj(
    const float* __restrict__ x, const int* __restrict__ mask,
    const float* __restrict__ td, const float* __restrict__ Wq,
    const float* __restrict__ bq, float* __restrict__ q)
{
  const int lane = threadIdx.x & 31;
  const int gw   = blockIdx.x * 8 + (threadIdx.x >> 5);  // 0..2047
  const int m0   = (gw >> 4) * 16;                        // 128 M-tiles
  const int n0   = (gw & 15) * 16;                        // 16 N-tiles
  const int half = lane >> 4;                             // K sub-pair select
  const int lm   = lane & 15;

  const int   mrow = m0 + lm;
  const float tdl  = td[mrow * TT + (TT - 1)];
  const float* arow = x  + ((size_t)mrow * TT + (TT - 1)) * DD;
  const float* brow = Wq + (size_t)(n0 + lm) * DD;        // B[kk,n] = Wq[n,kk]

  v8f c = {};
#pragma unroll 4
  for (int k0 = 0; k0 < DD; k0 += 4) {
    const int kk = k0 + 2 * half;
    v2f a = *(const v2f*)(arow + kk);
    a = a * tdl;                                           // fold time-decay
    v2f b = *(const v2f*)(brow + kk);
    c = __builtin_amdgcn_wmma_f32_16x16x4_f32(false, a, false, b,
                                              (short)0, c, false, false);
  }
  const int   n    = n0 + lm;
  const float bias = bq[n];
#pragma unroll
  for (int r = 0; r < 8; ++r) {
    const int row = m0 + r + 8 * half;                     // C VGPR layout
    float v = c[r] + bias;
    if (mask[row * TT + (TT - 1)] == 0) v = NEGV;          // q-side mask-fill
    q[(size_t)row * DD + n] = v;
  }
}

// ---------------------------------------------------------------------------
// Kernel 2: qk[b,:] = q[b,:] @ Wk   (qk[d] = sum_e q[e] * Wk[e,d])
// ---------------------------------------------------------------------------
__global__ __launch_bounds__(256) void k_qk(
    const float* __restrict__ q, const float* __restrict__ Wk,
    float* __restrict__ qk)
{
  const int lane = threadIdx.x & 31;
  const int gw   = blockIdx.x * 8 + (threadIdx.x >> 5);
  const int m0   = (gw >> 4) * 16;
  const int n0   = (gw & 15) * 16;
  const int half = lane >> 4;
  const int lm   = lane & 15;

  const float* arow = q + (size_t)(m0 + lm) * DD;
  v8f c = {};
#pragma unroll 4
  for (int k0 = 0; k0 < DD; k0 += 4) {
    const int kk = k0 + 2 * half;
    v2f a = *(const v2f*)(arow + kk);
    const float* bp = Wk + (size_t)kk * DD + (n0 + lm);    // B[kk,n]=Wk[kk,n]
    v2f b; b.x = bp[0]; b.y = bp[DD];
    c = __builtin_amdgcn_wmma_f32_16x16x4_f32(false, a, false, b,
                                              (short)0, c, false, false);
  }
  const int n = n0 + lm;
#pragma unroll
  for (int r = 0; r < 8; ++r)
    qk[(size_t)(m0 + r + 8 * half) * DD + n] = c[r];
}

// ---------------------------------------------------------------------------
// Kernel 3: per batch b — scores over s, softmax, xbar = sum_s p_s*td_s*x_s.
// Raw x[b] staged memory->LDS with GLOBAL_LOAD_ASYNC_TO_LDS_B128 (ASYNCcnt);
// the qbk/qsum reductions overlap the async DMA. td_s folded as a per-s
// scalar:  score = mask ? (td_s*dot(qk,x_s) + q.bk)/16 : (-1e5*sum(q))/16.
// xbar aliases the q buffer (q row fully consumed before the write).
// ---------------------------------------------------------------------------
__global__ __launch_bounds__(256) void k_attn(
    const float* __restrict__ x, const int* __restrict__ mask,
    const float* __restrict__ td, const float* __restrict__ qbuf,
    const float* __restrict__ qkbuf, const float* __restrict__ bk,
    float* __restrict__ xbar)
{
  __shared__ float Xs[TT * DD];     // raw x[b] tile (50 KB)
  __shared__ float qks[DD];
  __shared__ float tds[TT];
  __shared__ int   msk[TT];
  __shared__ float sc[TT];
  __shared__ float pv[TT];
  __shared__ float pw[TT];
  __shared__ float red1[8], red2[8];
  __shared__ float qbk_s, qsum_s;

  const int b = blockIdx.x;
  const int t = threadIdx.x;
  const int lane = t & 31, wave = t >> 5;

  // ---- issue async memory->LDS staging of x[b] (b128 per lane) ----
  const float* gsrc = x + (size_t)b * TT * DD;
#pragma unroll
  for (int i = t; i < (TT * DD) / 4; i += 256) {
    const unsigned loff = (unsigned)(uintptr_t)(&Xs[4 * i]); // LDS byte offset
    const float*   g    = gsrc + 4 * i;
    asm volatile("global_load_async_to_lds_b128 %0, %1, off"
                 :: "v"(loff), "v"(g) : "memory");
  }

  // ---- overlapped with DMA: stage scalars, reduce q.bk and sum(q) ----
  qks[t] = qkbuf[(size_t)b * DD + t];
  const float qv = qbuf[(size_t)b * DD + t];
  if (t < TT) { tds[t] = td[b * TT + t]; msk[t] = mask[b * TT + t]; }

  float v1 = qv * bk[t];
  float v2 = qv;
  for (int off = 16; off; off >>= 1) {
    v1 += __shfl_down(v1, off, 32);
    v2 += __shfl_down(v2, off, 32);
  }
  if (lane == 0) { red1[wave] = v1; red2[wave] = v2; }
  __syncthreads();
  if (t == 0) {
    float a = 0.f, s2 = 0.f;
    for (int i = 0; i < 8; ++i) { a += red1[i]; s2 += red2[i]; }
    qbk_s = a; qsum_s = s2;
  }

  // ---- DMA fence: own-wave ASYNCcnt, then cross-wave barrier ----
  asm volatile("s_wait_asynccnt 0x0" ::: "memory");
  __syncthreads();

  // ---- scores: each wave owns s = wave, wave+8, ... ----
  for (int s = wave; s < TT; s += 8) {
    float acc = 0.f;
#pragma unroll
    for (int j = 0; j < 8; ++j) {
      const int d = lane + 32 * j;
      acc += qks[d] * Xs[s * DD + d];
    }
    for (int off = 16; off; off >>= 1) acc += __shfl_down(acc, off, 32);
    if (lane == 0) {
      sc[s] = msk[s] ? (tds[s] * acc + qbk_s) * 0.0625f
                     : (NEGV * qsum_s) * 0.0625f;
    }
  }
  __syncthreads();

  // ---- softmax over 50; fold td_s into the value weights ----
  float mx = -3.402823466e38f;
  for (int s = 0; s < TT; ++s) mx = fmaxf(mx, sc[s]);
  if (t < TT) {
    const float e = __expf(sc[t] - mx);
    pv[t] = e;
    pw[t] = e * tds[t];
  }
  __syncthreads();
  float sum = 0.f;
  for (int s = 0; s < TT; ++s) sum += pv[s];
  const float inv = 1.0f / sum;

  float acc = 0.f;
  for (int s = 0; s < TT; ++s) acc += pw[s] * Xs[s * DD + t];
  xbar[(size_t)b * DD + t] = acc * inv;
}

// ---------------------------------------------------------------------------
// Kernel 4: out[b,:] = xbar[b,:] @ Wv^T + bv   (WMMA, same tiling as k1)
// ---------------------------------------------------------------------------
__global__ __launch_bounds__(256) void k_vproj(
    const float* __restrict__ xbar, const float* __restrict__ Wv,
    const float* __restrict__ bv, float* __restrict__ out)
{
  const int lane = threadIdx.x & 31;
  const int gw   = blockIdx.x * 8 + (threadIdx.x >> 5);
  const int m0   = (gw >> 4) * 16;
  const int n0   = (gw & 15) * 16;
  const int half = lane >> 4;
  const int lm   = lane & 15;

  const float* arow = xbar + (size_t)(m0 + lm) * DD;
  const float* brow = Wv   + (size_t)(n0 + lm) * DD;
  v8f c = {};
#pragma unroll 4
  for (int k0 = 0; k0 < DD; k0 += 4) {
    const int kk = k0 + 2 * half;
    v2f a = *(const v2f*)(arow + kk);
    v2f b = *(const v2f*)(brow + kk);
    c = __builtin_amdgcn_wmma_f32_16x16x4_f32(false, a, false, b,
                                              (short)0, c, false, false);
  }
  const int   n    = n0 + lm;
  const float bias = bv[n];
#pragma unroll
  for (int r = 0; r < 8; ++r)
    out[(size_t)(m0 + r + 8 * half) * DD + n] = c[r] + bias;
}

// ---------------------------------------------------------------------------
extern "C" void kernel_launch(void* const* d_in, const int* in_sizes, int n_in,
                              void* d_out, int out_size, void* d_ws, size_t ws_size,
                              hipStream_t stream) {
  const float* x    = (const float*)d_in[0];
  const int*   mask = (const int*)  d_in[1];
  const float* td   = (const float*)d_in[2];
  const float* Wq   = (const float*)d_in[3];
  const float* bq   = (const float*)d_in[4];
  const float* Wk   = (const float*)d_in[5];
  const float* bk   = (const float*)d_in[6];
  const float* Wv   = (const float*)d_in[7];
  const float* bv   = (const float*)d_in[8];
  float* out = (float*)d_out;

  // workspace: q (2 MB, reused as xbar) | qk (2 MB)
  float* qbuf  = (float*)d_ws;
  float* qkbuf = qbuf + (size_t)BB * DD;

  // 2048 waves (one 16x16 tile each) = 256 blocks of 8 waves
  k_qproj<<<dim3(256), dim3(256), 0, stream>>>(x, mask, td, Wq, bq, qbuf);
  k_qk   <<<dim3(256), dim3(256), 0, stream>>>(qbuf, Wk, qkbuf);
  k_attn <<<dim3(BB),  dim3(256), 0, stream>>>(x, mask, td, qbuf, qkbuf, bk, qbuf);
  k_vproj<<<dim3(256), dim3(256), 0, stream>>>(qbuf, Wv, bv, out);
}